// heat_conservation_17334488007105
// MI455X (gfx1250) — compile-verified
//
#include <hip/hip_runtime.h>

typedef float v2f __attribute__((ext_vector_type(2)));
typedef float v8f __attribute__((ext_vector_type(8)));

#define NX 200
#define PANEL_ROWS 16
#define NPANEL 13            // ceil(200/16)
#define LSTRIDE 204          // padded LDS row stride (floats): 16B-aligned rows, conflict-free
#define IMG (NX * NX)        // 40000
#define PANEL_FLOATS (PANEL_ROWS * NX)  // 3200
#define PANEL_SLOTS (PANEL_FLOATS / 4)  // 800 b128 slots
#define SLOTS_PER_WAVE (PANEL_SLOTS / 8) // 100 = 3*32 + 4

// gfx1250 async DMA: global memory -> LDS, tracked by ASYNCcnt (no VGPR round-trip).
// vdst = LDS byte address (low 32 bits of flat shared pointer), vaddr = 64-bit global addr.
__device__ __forceinline__ void async_copy_b128(unsigned lds_off, const float* gsrc) {
    asm volatile("global_load_async_to_lds_b128 %0, %1, off"
                 :: "v"(lds_off), "v"(gsrc)
                 : "memory");
}
__device__ __forceinline__ void wait_async_le8() {
    asm volatile("s_wait_asynccnt 0x8" ::: "memory");
}
__device__ __forceinline__ void wait_async_0() {
    asm volatile("s_wait_asynccnt 0x0" ::: "memory");
}

// Each wave issues exactly 4 async b128 ops covering slots [wv*100, wv*100+100) of one
// array's panel (100 = 3 full-EXEC + one 4-lane op -> ASYNCcnt += 4 per array, always).
// Global addresses are clamped in-bounds for the ragged last panel (garbage rows are
// zeroed after the wait, before compute).
__device__ __forceinline__ void issue_panel_array(const float* __restrict__ gsrc,
                                                  int gbase, int nval,
                                                  float* buf, int wv, int l) {
    const int s0 = wv * SLOTS_PER_WAVE;
    #pragma unroll
    for (int it = 0; it < 4; ++it) {
        if (it < 3 || l < 4) {
            const int s = s0 + it * 32 + l;
            const int e = s << 2;
            int eg = e;
            if (eg > nval - 4) eg = nval - 4;      // stay inside this image
            const int row = e / NX, col = e - row * NX;
            const unsigned loff =
                (unsigned)(unsigned long long)(buf + row * LSTRIDE + col);
            async_copy_b128(loff, gsrc + gbase + eg);
        }
    }
}

__global__ __launch_bounds__(256)
void heat_loss_batch_kernel(const float* __restrict__ heat,
                            const float* __restrict__ layout,
                            float* __restrict__ ws) {
    constexpr float COF = 0.25f * (0.1f / 199.0f) * (0.1f / 199.0f);
    const int b   = blockIdx.x;
    const int tid = threadIdx.x;
    const int l   = tid & 31;   // lane within wave32
    const int wv  = tid >> 5;   // wave id (8 waves)

    // double-buffered panels (async DMA writes next panel while WMMA reduces current)
    __shared__ __align__(16) float lds_h0[PANEL_ROWS * LSTRIDE + 16];
    __shared__ __align__(16) float lds_h1[PANEL_ROWS * LSTRIDE + 16];
    __shared__ __align__(16) float lds_l0[PANEL_ROWS * LSTRIDE + 16];
    __shared__ __align__(16) float lds_l1[PANEL_ROWS * LSTRIDE + 16];
    __shared__ float cs_h[NX], cs_l[NX], rs_h[NX], rs_l[NX];
    __shared__ float bh[4][NX];   // heat rows 0,1,198,199 (indexed by column)
    __shared__ float bc[4][NX];   // heat cols 0,1,198,199 (indexed by row)
    __shared__ float ra[256], rb[256];

    if (tid < NX) { cs_h[tid] = 0.f; cs_l[tid] = 0.f; rs_h[tid] = 0.f; rs_l[tid] = 0.f; }

    // prologue: start DMA of panel 0 into buffer 0
    issue_panel_array(heat,   b * IMG, PANEL_FLOATS, lds_h0, wv, l);
    issue_panel_array(layout, b * IMG, PANEL_FLOATS, lds_l0, wv, l);
    __syncthreads();   // covers the cs/rs zero-init visibility

    const v2f ones2 = {1.0f, 1.0f};

    for (int p = 0; p < NPANEL; ++p) {
        float* bufh = (p & 1) ? lds_h1 : lds_h0;
        float* bufl = (p & 1) ? lds_l1 : lds_l0;

        // ---- kick off next panel's DMA, then wait for current panel (<=8 in flight) ----
        if (p + 1 < NPANEL) {
            float* nh = ((p + 1) & 1) ? lds_h1 : lds_h0;
            float* nl = ((p + 1) & 1) ? lds_l1 : lds_l0;
            const int nb = b * IMG + (p + 1) * PANEL_FLOATS;
            int nv = IMG - (p + 1) * PANEL_FLOATS;
            if (nv > PANEL_FLOATS) nv = PANEL_FLOATS;
            issue_panel_array(heat,   nb, nv, nh, wv, l);
            issue_panel_array(layout, nb, nv, nl, wv, l);
            wait_async_le8();   // the 8 oldest (panel p) are done; panel p+1 streams on
        } else {
            wait_async_0();
        }
        __syncthreads();

        if (p == NPANEL - 1) {
            // last panel holds only 8 valid rows; zero rows 8..15 so the all-ones
            // column-sum WMMAs accumulate zeros there
            const float4 z4 = make_float4(0.f, 0.f, 0.f, 0.f);
            for (int s = tid; s < (8 * LSTRIDE) / 4; s += 256) {
                const int idx = 8 * LSTRIDE + s * 4;
                *(float4*)&bufh[idx] = z4;
                *(float4*)&bufl[idx] = z4;
            }
            __syncthreads();
        }

        // ---- capture reflect-pad boundary values of heat ----
        if (p == 0 && tid < NX) {
            bh[0][tid] = bufh[tid];
            bh[1][tid] = bufh[LSTRIDE + tid];
        }
        if (p == NPANEL - 1 && tid < NX) {   // global rows 198,199 = panel rows 6,7
            bh[2][tid] = bufh[6 * LSTRIDE + tid];
            bh[3][tid] = bufh[7 * LSTRIDE + tid];
        }
        if (tid < 64) {
            const int r = tid & 15, w = tid >> 4;
            const int gr = p * PANEL_ROWS + r;
            if (gr < NX) {
                const int c = (w == 0) ? 0 : (w == 1) ? 1 : (w == 2) ? 198 : 199;
                bc[w][gr] = bufh[r * LSTRIDE + c];
            }
        }

        // ---- WMMA reduction tasks (wave-uniform branching; EXEC all-ones at wmma) ----
        const int arr = wv >> 2;                  // 0: heat, 1: layout
        const float* P = arr ? bufl : bufh;
        float* CS = arr ? cs_l : cs_h;
        float* RS = arr ? rs_l : rs_h;
        const int q  = wv & 3;
        const int r0 = p * PANEL_ROWS;

        if (q < 2) {
            // partial ROW sums: D += A(H 16x4) x Ones(4x16), K-chunks [q*25, q*25+25)
            v8f acc = {0.f, 0.f, 0.f, 0.f, 0.f, 0.f, 0.f, 0.f};
            const int row = l & 15, half = l >> 4;
            const float* abase = P + row * LSTRIDE + half * 2;
            const int kk0 = q * 25, kk1 = kk0 + 25;
            for (int kk = kk0; kk < kk1; ++kk) {
                v2f a = *(const v2f*)(abase + kk * 4);   // ds_load_b64: K pair per lane-half
                acc = __builtin_amdgcn_wmma_f32_16x16x4_f32(
                          false, a, false, ones2, (short)0, acc, false, false);
            }
            // D[m,0]: lane 0 holds M=v in VGPR v; lane 16 holds M=v+8
            if ((l & 15) == 0) {
                const int mb = (l >> 4) * 8;
                #pragma unroll
                for (int v = 0; v < 8; ++v) {
                    const int gr = r0 + mb + v;
                    if (gr < NX) atomicAdd(&RS[gr], acc[v]);  // ds_add_f32; 2 commutative adds
                }
            }
        } else {
            // partial COLUMN sums: D += Ones(16x4) x B(H 4x16), strips [s0,s1)
            const int s0 = (q == 2) ? 0 : 7;
            const int s1 = (q == 2) ? 7 : 13;
            const int n = l & 15, half = l >> 4;
            for (int s = s0; s < s1; ++s) {
                const int c0 = s * 16;
                v8f acc = {0.f, 0.f, 0.f, 0.f, 0.f, 0.f, 0.f, 0.f};
                for (int kk = 0; kk < 4; ++kk) {
                    const float* pc = P + (kk * 4 + half) * LSTRIDE + c0 + n;
                    v2f bb = { pc[0], pc[2 * LSTRIDE] };  // rows 4kk..4kk+3, each exactly once
                    acc = __builtin_amdgcn_wmma_f32_16x16x4_f32(
                              false, ones2, false, bb, (short)0, acc, false, false);
                }
                if (l < 16) {   // D[0,n] lives in VGPR0, lanes 0..15
                    const int c = c0 + l;
                    if (c < NX) atomicAdd(&CS[c], acc[0]);
                }
            }
        }
        __syncthreads();
    }

    // ---- per-column (vertical) and per-row (horizontal) diffs from sums ----
    float dv = 0.f, dh = 0.f;
    if (tid < NX) {
        const int j = tid;
        const float S  = cs_h[j];
        const float SW = cs_h[(j >= 1) ? j - 1 : 1];
        const float SE = cs_h[(j <= NX - 2) ? j + 1 : NX - 2];
        const float nv = 0.25f * (2.f * S + bh[1][j] - bh[3][j] - bh[0][j] + bh[2][j] + SW + SE)
                       + COF * cs_l[j] - S;
        dv = fabsf(nv) * (1.0f / (float)NX);

        const int i = tid;
        const float R   = rs_h[i];
        const float RN  = rs_h[(i >= 1) ? i - 1 : 1];
        const float RS2 = rs_h[(i <= NX - 2) ? i + 1 : NX - 2];
        const float nh = 0.25f * (2.f * R + bc[1][i] - bc[3][i] - bc[0][i] + bc[2][i] + RN + RS2)
                       + COF * rs_l[i] - R;
        dh = fabsf(nh) * (1.0f / (float)NX);
    }

    // ---- OHEM: per-batch min/max normalization + weighted sum (deterministic trees) ----
    float partial = 0.f;
    for (int pass = 0; pass < 2; ++pass) {
        const float d = pass ? dh : dv;
        ra[tid] = (tid < NX) ? d :  3.4e38f;
        rb[tid] = (tid < NX) ? d : -3.4e38f;
        __syncthreads();
        for (int off = 128; off > 0; off >>= 1) {
            if (tid < off) {
                ra[tid] = fminf(ra[tid], ra[tid + off]);
                rb[tid] = fmaxf(rb[tid], rb[tid + off]);
            }
            __syncthreads();
        }
        const float mn = ra[0], mx = rb[0];
        __syncthreads();
        ra[tid] = (tid < NX) ? (10.f * (d - mn) / (mx - mn)) * d : 0.f;
        __syncthreads();
        for (int off = 128; off > 0; off >>= 1) {
            if (tid < off) ra[tid] += ra[tid + off];
            __syncthreads();
        }
        partial += ra[0];
        __syncthreads();
    }
    if (tid == 0) ws[b] = partial;
}

__global__ __launch_bounds__(256)
void heat_loss_finalize_kernel(const float* __restrict__ ws, float* __restrict__ out, int B) {
    __shared__ float r[256];
    float s = 0.f;
    for (int i = threadIdx.x; i < B; i += 256) s += ws[i];
    r[threadIdx.x] = s;
    __syncthreads();
    for (int off = 128; off > 0; off >>= 1) {
        if (threadIdx.x < off) r[threadIdx.x] += r[threadIdx.x + off];
        __syncthreads();
    }
    if (threadIdx.x == 0) out[0] = r[0] * (1.0f / ((float)B * (float)NX));
}

extern "C" void kernel_launch(void* const* d_in, const int* in_sizes, int n_in,
                              void* d_out, int out_size, void* d_ws, size_t ws_size,
                              hipStream_t stream) {
    (void)n_in; (void)out_size; (void)ws_size;
    const float* layout = (const float*)d_in[0];  // setup_inputs order: layout, heat
    const float* heat   = (const float*)d_in[1];
    const int B = in_sizes[0] / IMG;
    float* ws = (float*)d_ws;

    hipLaunchKernelGGL(heat_loss_batch_kernel, dim3(B), dim3(256), 0, stream,
                       heat, layout, ws);
    hipLaunchKernelGGL(heat_loss_finalize_kernel, dim3(1), dim3(256), 0, stream,
                       ws, (float*)d_out, B);
}